// PatchNet_53223234732617
// MI455X (gfx1250) — compile-verified
//
#include <hip/hip_runtime.h>
#include <hip/hip_bf16.h>

typedef __attribute__((ext_vector_type(2))) float v2f;
typedef __attribute__((ext_vector_type(8))) float v8f;

#define T_BATCH 128
#define N_TOK 256
#define C_DIM 1408
#define CN1 16
#define CN2 6
#define KNN1 8
#define KNN2 3
#define INV_SQRT_C (1.0f / 37.523326f) /* 1/sqrt(1408) */

// ---------------------------------------------------------------------------
// K0: per-row squared norms  sq[t*N+i] = sum_k x^2
// ---------------------------------------------------------------------------
__global__ __launch_bounds__(256) void k_rowsq(const float* __restrict__ x,
                                               float* __restrict__ sq) {
  int row  = blockIdx.x * 8 + (threadIdx.x >> 5);
  int lane = threadIdx.x & 31;
  const float* p = x + (size_t)row * C_DIM;
  float s = 0.0f;
  for (int k = lane; k < C_DIM; k += 32) { float v = p[k]; s += v * v; }
  for (int off = 16; off; off >>= 1) s += __shfl_xor(s, off, 32);
  if (lane == 0) sq[row] = s;
}

// ---------------------------------------------------------------------------
// K1: Gram via V_WMMA_F32_16X16X4_F32 -> dist[t][i][j]
// One block = one 128x128 tile of the 256x256 matrix. 8 waves, each 32x64.
// ---------------------------------------------------------------------------
__global__ __launch_bounds__(256) void k_gram_dist(const float* __restrict__ x,
                                                   const float* __restrict__ sq,
                                                   float* __restrict__ dist) {
  __shared__ float As[128 * 34];   // stride 34: conflict-free + 8B aligned pairs
  __shared__ float Bs[128 * 34];
  __shared__ float sqA[128];
  __shared__ float sqB[128];

  int b = blockIdx.x;
  int t = b >> 2;
  int rowBase = ((b >> 1) & 1) * 128;
  int colBase = (b & 1) * 128;
  int tid  = threadIdx.x;
  int lane = tid & 31;
  int w    = tid >> 5;
  int wm   = w >> 1;          // 0..3 -> 32-row slab
  int wn   = w & 1;           // 0..1 -> 64-col slab
  int lm   = lane & 15;
  int kk   = (lane >> 4) * 2;
  int hb   = lane >> 4;

  if (tid < 128) sqA[tid] = sq[t * N_TOK + rowBase + tid];
  else           sqB[tid - 128] = sq[t * N_TOK + colBase + (tid - 128)];

  const float* xt = x + (size_t)t * N_TOK * C_DIM;

  v8f vzero = {0.f,0.f,0.f,0.f,0.f,0.f,0.f,0.f};
  v8f acc[2][4];
  for (int mi = 0; mi < 2; ++mi)
    for (int ni = 0; ni < 4; ++ni) acc[mi][ni] = vzero;

  for (int k0 = 0; k0 < C_DIM; k0 += 32) {
    __syncthreads();
    for (int rr = w; rr < 128; rr += 8) {
      As[rr * 34 + lane] = xt[(size_t)(rowBase + rr) * C_DIM + k0 + lane];
      Bs[rr * 34 + lane] = xt[(size_t)(colBase + rr) * C_DIM + k0 + lane];
      if (k0 + 32 < C_DIM) {
        __builtin_prefetch(&xt[(size_t)(rowBase + rr) * C_DIM + k0 + 32 + lane], 0, 1);
        __builtin_prefetch(&xt[(size_t)(colBase + rr) * C_DIM + k0 + 32 + lane], 0, 1);
      }
    }
    __syncthreads();

    for (int k = 0; k < 32; k += 4) {
      v2f bfrag[4], afrag[2];
      for (int ni = 0; ni < 4; ++ni) {
        int base = (wn * 64 + ni * 16 + lm) * 34 + k + kk;
        bfrag[ni].x = Bs[base];
        bfrag[ni].y = Bs[base + 1];
      }
      for (int mi = 0; mi < 2; ++mi) {
        int base = (wm * 32 + mi * 16 + lm) * 34 + k + kk;
        afrag[mi].x = As[base];
        afrag[mi].y = As[base + 1];
      }
      for (int mi = 0; mi < 2; ++mi)
        for (int ni = 0; ni < 4; ++ni)
          acc[mi][ni] = __builtin_amdgcn_wmma_f32_16x16x4_f32(
              false, afrag[mi], false, bfrag[ni], (short)0, acc[mi][ni],
              false, false);
    }
  }

  // epilogue: dist = sqrt(max(sq_i + sq_j - 2G, 0)) / sqrt(C)
  for (int mi = 0; mi < 2; ++mi)
    for (int ni = 0; ni < 4; ++ni)
      for (int v = 0; v < 8; ++v) {
        int i = wm * 32 + mi * 16 + v + 8 * hb;   // local row 0..127
        int j = wn * 64 + ni * 16 + lm;           // local col 0..127
        float g  = acc[mi][ni][v];
        float d2 = sqA[i] + sqB[j] - 2.0f * g;
        float d  = sqrtf(fmaxf(d2, 0.0f)) * INV_SQRT_C;
        dist[((size_t)t * N_TOK + rowBase + i) * N_TOK + colBase + j] = d;
      }
}

// ---------------------------------------------------------------------------
// K2: per-row kNN(8) density + row max. One wave per row.
// ---------------------------------------------------------------------------
__global__ __launch_bounds__(256) void k_knn_density(const float* __restrict__ dist,
                                                     float* __restrict__ dens,
                                                     float* __restrict__ rowmax) {
  int row  = blockIdx.x * 8 + (threadIdx.x >> 5);
  int lane = threadIdx.x & 31;
  const float* dr = dist + (size_t)row * N_TOK;

  float vals[8];
  float rmax = 0.0f;
  for (int p = 0; p < 8; ++p) {
    vals[p] = dr[lane + 32 * p];
    rmax = fmaxf(rmax, vals[p]);
  }
  for (int off = 16; off; off >>= 1) rmax = fmaxf(rmax, __shfl_xor(rmax, off, 32));

  float sumsq = 0.0f;
  for (int it = 0; it < KNN1; ++it) {
    float lmin = 1e30f;
    for (int p = 0; p < 8; ++p) lmin = fminf(lmin, vals[p]);
    float gmin = lmin;
    for (int off = 16; off; off >>= 1) gmin = fminf(gmin, __shfl_xor(gmin, off, 32));
    sumsq += gmin * gmin;
    unsigned long long bal = __ballot(lmin == gmin);
    int owner = __ffsll(bal) - 1;
    if (lane == owner) {
      for (int p = 0; p < 8; ++p)
        if (vals[p] == gmin) { vals[p] = 1e30f; break; }
    }
  }
  if (lane == 0) {
    dens[row]   = expf(-sumsq * (1.0f / (float)KNN1));
    rowmax[row] = rmax;
  }
}

// ---------------------------------------------------------------------------
// K3: parent-dist, score, top-16 centers, nearest-center assignment. 1 blk/batch.
// ---------------------------------------------------------------------------
__global__ __launch_bounds__(256) void k_parent_assign(const float* __restrict__ dist,
                                                       const float* __restrict__ dens,
                                                       const float* __restrict__ rowmax,
                                                       int* __restrict__ idx1) {
  __shared__ float densL[256];
  __shared__ float scoreL[256];
  __shared__ int   downL[CN1];
  __shared__ float red[8];
  __shared__ float dmaxL;

  int t = blockIdx.x, tid = threadIdx.x, lane = tid & 31, w = tid >> 5;
  densL[tid] = dens[t * N_TOK + tid];

  float rm = rowmax[t * N_TOK + tid];
  for (int off = 16; off; off >>= 1) rm = fmaxf(rm, __shfl_xor(rm, off, 32));
  if (lane == 0) red[w] = rm;
  __syncthreads();
  if (tid == 0) {
    float m = red[0];
    for (int i = 1; i < 8; ++i) m = fmaxf(m, red[i]);
    dmaxL = m;
  }
  __syncthreads();
  float dmax = dmaxL;
  const float* dt = dist + (size_t)t * N_TOK * N_TOK;

  for (int r = 0; r < 32; ++r) {
    int i = w * 32 + r;
    float di = densL[i];
    float m = 1e30f;
    for (int p = 0; p < 8; ++p) {
      int j = lane + 32 * p;
      float d = dt[(size_t)i * N_TOK + j];
      m = fminf(m, (densL[j] > di) ? d : dmax);
    }
    for (int off = 16; off; off >>= 1) m = fminf(m, __shfl_xor(m, off, 32));
    if (lane == 0) scoreL[i] = m * di;
  }
  __syncthreads();

  if (w == 0) {  // top-16 by score, descending
    for (int c = 0; c < CN1; ++c) {
      float lb = -1e30f; int li = 0;
      for (int p = 0; p < 8; ++p) {
        int i = lane + 32 * p;
        float s = scoreL[i];
        if (s > lb) { lb = s; li = i; }
      }
      float gb = lb;
      for (int off = 16; off; off >>= 1) gb = fmaxf(gb, __shfl_xor(gb, off, 32));
      unsigned long long bal = __ballot(lb == gb);
      int owner = __ffsll(bal) - 1;
      int sel = __shfl(li, owner, 32);
      if (lane == owner) scoreL[sel] = -1e30f;
      if (lane == 0) downL[c] = sel;
      __threadfence_block();
    }
  }
  __syncthreads();

  {
    int i = tid;
    float best = 1e30f; int bc = 0;
    for (int c = 0; c < CN1; ++c) {
      float d = dt[(size_t)downL[c] * N_TOK + i];
      if (d < best) { best = d; bc = c; }
    }
    for (int c = 0; c < CN1; ++c) if (downL[c] == i) bc = c;
    idx1[t * N_TOK + i] = bc;
  }
}

// ---------------------------------------------------------------------------
// K5: merge1 = scatter-mean of 256 tokens into 16 clusters (per channel chunk)
// ---------------------------------------------------------------------------
__global__ __launch_bounds__(256) void k_merge1(const float* __restrict__ x,
                                                const int* __restrict__ idx1,
                                                float* __restrict__ meta1) {
  __shared__ int   cidx[256];
  __shared__ int   cntI[CN1];
  __shared__ float accL[CN1 * 256];
  int t = blockIdx.x, chunk = blockIdx.y, tid = threadIdx.x;
  int ch = chunk * 256 + tid;

  cidx[tid] = idx1[t * N_TOK + tid];
  if (tid < CN1) cntI[tid] = 0;
  for (int c = 0; c < CN1; ++c) accL[c * 256 + tid] = 0.0f;
  __syncthreads();
  atomicAdd(&cntI[cidx[tid]], 1);
  __syncthreads();

  bool ok = ch < C_DIM;
  const float* xt = x + (size_t)t * N_TOK * C_DIM;
  for (int i = 0; i < N_TOK; ++i) {
    int c = cidx[i];
    if (ok) accL[c * 256 + tid] += xt[(size_t)i * C_DIM + ch];
  }
  __syncthreads();
  if (ok)
    for (int c = 0; c < CN1; ++c) {
      float wgt = 1.0f / ((float)cntI[c] + 1e-6f);
      meta1[((size_t)t * CN1 + c) * C_DIM + ch] = accL[c * 256 + tid] * wgt;
    }
}

// ---------------------------------------------------------------------------
// K6: layer-2 DPC-KNN in one block per batch. Gram = one WMMA tile (A==B).
// 8 waves split K=1408 into 176-wide slices, LDS-reduce, then small serial DPC.
// ---------------------------------------------------------------------------
__global__ __launch_bounds__(256) void k_layer2(const float* __restrict__ meta1,
                                                int* __restrict__ idx2) {
  __shared__ float part[8][256];
  __shared__ float g2[256];
  __shared__ float d2m[256];
  __shared__ float dens2[CN1];
  __shared__ float rmax2[CN1];
  __shared__ float score2[CN1];
  __shared__ int   down2[CN2];
  __shared__ float dmax2;

  int t = blockIdx.x, tid = threadIdx.x, lane = tid & 31, w = tid >> 5;
  const float* mt = meta1 + (size_t)t * CN1 * C_DIM;
  int lm = lane & 15, kk = (lane >> 4) * 2;

  v8f acc = {0.f,0.f,0.f,0.f,0.f,0.f,0.f,0.f};
  int kbeg = w * 176, kend = kbeg + 176;
  for (int k = kbeg; k < kend; k += 4) {
    v2f a;
    a.x = mt[(size_t)lm * C_DIM + k + kk];
    a.y = mt[(size_t)lm * C_DIM + k + kk + 1];
    // Gram: B-fragment of X^T equals A-fragment of X under CDNA5 layouts
    acc = __builtin_amdgcn_wmma_f32_16x16x4_f32(false, a, false, a, (short)0,
                                                acc, false, false);
  }
  for (int v = 0; v < 8; ++v) part[w][v * 32 + lane] = acc[v];
  __syncthreads();

  if (tid < 32) {
    for (int v = 0; v < 8; ++v) {
      float s = 0.0f;
      for (int ww = 0; ww < 8; ++ww) s += part[ww][v * 32 + lane];
      int row = v + 8 * (lane >> 4);
      int col = lane & 15;
      g2[row * 16 + col] = s;
    }
  }
  __syncthreads();
  {
    int i = tid >> 4, j = tid & 15;
    float gij = g2[i * 16 + j];
    float dd  = g2[i * 16 + i] + g2[j * 16 + j] - 2.0f * gij;
    d2m[i * 16 + j] = sqrtf(fmaxf(dd, 0.0f)) * INV_SQRT_C;
  }
  __syncthreads();
  if (tid < CN1) {  // kNN k=3 (includes self=0) + row max
    float m0 = 1e30f, m1 = 1e30f, m2 = 1e30f, rmx = 0.0f;
    for (int j = 0; j < CN1; ++j) {
      float d = d2m[tid * 16 + j];
      rmx = fmaxf(rmx, d);
      if (d < m0)      { m2 = m1; m1 = m0; m0 = d; }
      else if (d < m1) { m2 = m1; m1 = d; }
      else if (d < m2) { m2 = d; }
    }
    dens2[tid] = expf(-(m0*m0 + m1*m1 + m2*m2) * (1.0f / (float)KNN2));
    rmax2[tid] = rmx;
  }
  __syncthreads();
  if (tid == 0) {
    float m = 0.0f;
    for (int i = 0; i < CN1; ++i) m = fmaxf(m, rmax2[i]);
    dmax2 = m;
  }
  __syncthreads();
  if (tid < CN1) {
    float di = dens2[tid];
    float m = 1e30f;
    for (int j = 0; j < CN1; ++j) {
      float d = d2m[tid * 16 + j];
      m = fminf(m, (dens2[j] > di) ? d : dmax2);
    }
    score2[tid] = m * di;
  }
  __syncthreads();
  if (tid == 0) {
    for (int c = 0; c < CN2; ++c) {
      float b = -1e30f; int bi = 0;
      for (int i = 0; i < CN1; ++i)
        if (score2[i] > b) { b = score2[i]; bi = i; }
      down2[c] = bi;
      score2[bi] = -1e30f;
    }
  }
  __syncthreads();
  if (tid < CN1) {
    float best = 1e30f; int bc = 0;
    for (int c = 0; c < CN2; ++c) {
      float d = d2m[down2[c] * 16 + tid];
      if (d < best) { best = d; bc = c; }
    }
    for (int c = 0; c < CN2; ++c) if (down2[c] == tid) bc = c;
    idx2[t * CN1 + tid] = bc;
  }
}

// ---------------------------------------------------------------------------
// K7: merge2 = scatter-mean of 16 meta1 tokens into 6 clusters
// ---------------------------------------------------------------------------
__global__ __launch_bounds__(256) void k_merge2(const float* __restrict__ meta1,
                                                const int* __restrict__ idx2,
                                                float* __restrict__ meta2) {
  __shared__ int   cidx[CN1];
  __shared__ int   cntI[CN2];
  __shared__ float accL[CN2 * 256];
  int t = blockIdx.x, chunk = blockIdx.y, tid = threadIdx.x;
  int ch = chunk * 256 + tid;

  if (tid < CN1) cidx[tid] = idx2[t * CN1 + tid];
  if (tid < CN2) cntI[tid] = 0;
  for (int c = 0; c < CN2; ++c) accL[c * 256 + tid] = 0.0f;
  __syncthreads();
  if (tid < CN1) atomicAdd(&cntI[cidx[tid]], 1);
  __syncthreads();

  bool ok = ch < C_DIM;
  const float* mt = meta1 + (size_t)t * CN1 * C_DIM;
  for (int i = 0; i < CN1; ++i) {
    int c = cidx[i];
    if (ok) accL[c * 256 + tid] += mt[(size_t)i * C_DIM + ch];
  }
  __syncthreads();
  if (ok)
    for (int c = 0; c < CN2; ++c)
      meta2[((size_t)t * CN2 + c) * C_DIM + ch] =
          accL[c * 256 + tid] * (1.0f / ((float)cntI[c] + 1e-6f));
}

// ---------------------------------------------------------------------------
// K8: softmax modulation + stable regroup of [meta2;meta1] -> out [T,22,C]
// ---------------------------------------------------------------------------
__global__ __launch_bounds__(256) void k_group(const float* __restrict__ meta1,
                                               const float* __restrict__ meta2,
                                               const int* __restrict__ idx2,
                                               const float* __restrict__ sw,
                                               const float* __restrict__ sb,
                                               float* __restrict__ out) {
  __shared__ float rowmean[CN2];
  __shared__ float modu[CN2];
  __shared__ int   keys[CN2 + CN1];
  __shared__ int   srcOf[CN2 + CN1];
  __shared__ float scaleOf[CN2 + CN1];

  int t = blockIdx.x, tid = threadIdx.x, lane = tid & 31, w = tid >> 5;
  const float* m2 = meta2 + (size_t)t * CN2 * C_DIM;

  if (w < CN2) {
    float s = 0.0f;
    for (int k = lane; k < C_DIM; k += 32) s += m2[(size_t)w * C_DIM + k];
    for (int off = 16; off; off >>= 1) s += __shfl_xor(s, off, 32);
    if (lane == 0) rowmean[w] = s * (1.0f / (float)C_DIM);
  }
  __syncthreads();
  if (tid == 0) {
    float lg[CN2], mx = -1e30f;
    for (int r = 0; r < CN2; ++r) {
      float a = sb[r];
      for (int j = 0; j < CN2; ++j) a += rowmean[j] * sw[r * CN2 + j];
      lg[r] = a; mx = fmaxf(mx, a);
    }
    float sum = 0.0f;
    for (int r = 0; r < CN2; ++r) { lg[r] = expf(lg[r] - mx); sum += lg[r]; }
    for (int r = 0; r < CN2; ++r) modu[r] = lg[r] / sum;
  }
  __syncthreads();
  if (tid < CN2 + CN1) {
    int cid, pos;
    if (tid < CN2) { cid = tid; pos = 0; }
    else { cid = idx2[t * CN1 + (tid - CN2)]; pos = tid - CN2 + 1; }
    keys[tid] = cid * (CN1 + 2) + pos;       // unique sort keys
  }
  __syncthreads();
  if (tid < CN2 + CN1) {
    int myk = keys[tid], rank = 0;
    for (int b2 = 0; b2 < CN2 + CN1; ++b2) rank += (keys[b2] < myk);
    srcOf[rank] = tid;
    scaleOf[rank] = modu[myk / (CN1 + 2)];
  }
  __syncthreads();

  const float* m1 = meta1 + (size_t)t * CN1 * C_DIM;
  float* ot = out + (size_t)t * (CN2 + CN1) * C_DIM;
  for (int s = 0; s < CN2 + CN1; ++s) {
    int src = srcOf[s];
    float sc = scaleOf[s];
    const float* p = (src < CN2) ? (m2 + (size_t)src * C_DIM)
                                 : (m1 + (size_t)(src - CN2) * C_DIM);
    for (int ch = tid; ch < C_DIM; ch += 256)
      ot[(size_t)s * C_DIM + ch] = p[ch] * sc;
  }
}

// ---------------------------------------------------------------------------
extern "C" void kernel_launch(void* const* d_in, const int* in_sizes, int n_in,
                              void* d_out, int out_size, void* d_ws, size_t ws_size,
                              hipStream_t stream) {
  (void)in_sizes; (void)n_in; (void)out_size; (void)ws_size;
  const float* x  = (const float*)d_in[0];
  const float* sw = (const float*)d_in[1];
  const float* sb = (const float*)d_in[2];
  float* out = (float*)d_out;

  char* ws = (char*)d_ws;
  size_t off = 0;
  auto take = [&](size_t elems) -> char* {
    char* p = ws + off;
    off += (elems * 4 + 255) & ~(size_t)255;
    return p;
  };
  float* sq     = (float*)take((size_t)T_BATCH * N_TOK);
  float* dist1  = (float*)take((size_t)T_BATCH * N_TOK * N_TOK);   // 33.5 MB
  float* dens   = (float*)take((size_t)T_BATCH * N_TOK);
  float* rowmax = (float*)take((size_t)T_BATCH * N_TOK);
  int*   idx1   = (int*)  take((size_t)T_BATCH * N_TOK);
  float* meta1  = (float*)take((size_t)T_BATCH * CN1 * C_DIM);     // 11.5 MB
  int*   idx2   = (int*)  take((size_t)T_BATCH * CN1);
  float* meta2  = (float*)take((size_t)T_BATCH * CN2 * C_DIM);     // 4.3 MB

  k_rowsq        <<<(T_BATCH * N_TOK) / 8, 256, 0, stream>>>(x, sq);
  k_gram_dist    <<<T_BATCH * 4,           256, 0, stream>>>(x, sq, dist1);
  k_knn_density  <<<(T_BATCH * N_TOK) / 8, 256, 0, stream>>>(dist1, dens, rowmax);
  k_parent_assign<<<T_BATCH,               256, 0, stream>>>(dist1, dens, rowmax, idx1);
  k_merge1       <<<dim3(T_BATCH, 6),      256, 0, stream>>>(x, idx1, meta1);
  k_layer2       <<<T_BATCH,               256, 0, stream>>>(meta1, idx2);
  k_merge2       <<<dim3(T_BATCH, 6),      256, 0, stream>>>(meta1, idx2, meta2);
  k_group        <<<T_BATCH,               256, 0, stream>>>(meta1, meta2, idx2, sw, sb, out);
}